// allen_cahn_gradient_free_2628519985937
// MI455X (gfx1250) — compile-verified
//
#include <hip/hip_runtime.h>
#include <math.h>

// CDNA5 / gfx1250, wave32 only.
typedef __attribute__((ext_vector_type(2))) float v2f;
typedef __attribute__((ext_vector_type(8))) float v8f;

#define N_F     12288
#define RADIUS  0.02f
#define BLOCK   64            // 2 wave32 per block
#define GRID    (N_F / BLOCK) // 192 blocks
#define INFF    __builtin_inff()

__global__ __launch_bounds__(BLOCK)
void allen_cahn_knn_inv_kernel(const float* __restrict__ x, float* __restrict__ out)
{
    __shared__ float xs[N_F];     // 48 KB: entire point set, L2->LDS once per block
    __shared__ float ssum[BLOCK]; // per-row sum(dx^2) extracted from WMMA C tile

    const int tid = threadIdx.x;
    const int row = blockIdx.x * BLOCK + tid;

    // ---- Stage all x into LDS (vectorized, coalesced) ----
    {
        const float4* g4 = reinterpret_cast<const float4*>(x);
        float4*       s4 = reinterpret_cast<float4*>(xs);
        for (int t = tid; t < N_F / 4; t += BLOCK) s4[t] = g4[t];
    }
    __syncthreads();

    const float xi = xs[row];

    // ---- Brute-force top-4 smallest |x_j - x_i| within RADIUS, j != i ----
    // Uniform LDS addresses across the wave -> bank-broadcast ds_load_b128.
    float d0 = INFF, d1 = INFF, d2 = INFF, d3 = INFF;
    int   i0 = -1,   i1 = -1,   i2 = -1,   i3 = -1;

    const float4* xs4 = reinterpret_cast<const float4*>(xs);
    for (int j4 = 0; j4 < N_F / 4; ++j4) {
        const float4 v  = xs4[j4];
        const int    jb = j4 << 2;
        #pragma unroll
        for (int u = 0; u < 4; ++u) {
            const float xj = (u == 0) ? v.x : (u == 1) ? v.y : (u == 2) ? v.z : v.w;
            const int   j  = jb + u;
            float d = fabsf(xj - xi);
            const bool ok = (d <= RADIUS) && (j != row);
            d = ok ? d : INFF;
            // stable bubble insert (strict <) == jax.lax.top_k tie order
            if (d < d3) {
                d3 = d; i3 = j;
                if (d3 < d2) {
                    float td = d3; d3 = d2; d2 = td; int ti = i3; i3 = i2; i2 = ti;
                    if (d2 < d1) {
                        td = d2; d2 = d1; d1 = td; ti = i2; i2 = i1; i1 = ti;
                        if (d1 < d0) {
                            td = d1; d1 = d0; d0 = td; ti = i1; i1 = i0; i0 = ti;
                        }
                    }
                }
            }
        }
    }

    // invalid slots -> self index (dx contribution becomes exactly 0)
    if (i0 < 0) i0 = row;
    if (i1 < 0) i1 = row;
    if (i2 < 0) i2 = row;
    if (i3 < 0) i3 = row;

    const float dx0 = xs[i0] - xi, dx1 = xs[i1] - xi, dx2 = xs[i2] - xi, dx3 = xs[i3] - xi;
    const float q0 = dx0 * dx0, q1 = dx1 * dx1, q2 = dx2 * dx2, q3 = dx3 * dx3;

    // ---- sum_k dx_k^2 for 16 rows at a time via V_WMMA_F32_16X16X4_F32 ----
    // D = A(16x4 = dx^2) * B(4x16 = ones) + 0  ->  D[n, :] = sum_k dx^2[n,k]
    // B == all-ones is layout-invariant; A uses the documented 32-bit A 16x4
    // layout: lanes 0-15 hold (K=0,K=1), lanes 16-31 hold (K=2,K=3).
    const int lane = tid & 31;
    const int wave = tid >> 5;

    v2f bones; bones.x = 1.0f; bones.y = 1.0f;
    v8f czero = {};

    #pragma unroll
    for (int h = 0; h < 2; ++h) {          // half-wave h covers wave rows h*16..h*16+15
        const int src = (h << 4) | (lane & 15);
        const float s0 = __shfl(q0, src, 32);
        const float s1 = __shfl(q1, src, 32);
        const float s2 = __shfl(q2, src, 32);
        const float s3 = __shfl(q3, src, 32);
        v2f a;
        a.x = (lane < 16) ? s0 : s2;
        a.y = (lane < 16) ? s1 : s3;

        v8f dacc = __builtin_amdgcn_wmma_f32_16x16x4_f32(
            /*neg_a=*/false, a, /*neg_b=*/false, bones,
            /*c_mod=*/(short)0, czero, /*reuse_a=*/false, /*reuse_b=*/false);

        // C layout: VGPR v -> (M=v, N=lane) for lanes 0-15, (M=v+8, N=lane-16)
        // for lanes 16-31. Column N==0 lives entirely in lanes 0 and 16.
        if (lane == 0) {
            #pragma unroll
            for (int v = 0; v < 8; ++v) ssum[(wave << 5) + (h << 4) + v] = dacc[v];
        }
        if (lane == 16) {
            #pragma unroll
            for (int v = 0; v < 8; ++v) ssum[(wave << 5) + (h << 4) + 8 + v] = dacc[v];
        }
    }
    __syncthreads();

    const float sumsq = ssum[tid];

    // ---- Outputs: invs [N] then zn [N,4] (as f32; indices < 2^24 exact) ----
    out[row] = 1.0f / (sumsq + 1e-8f);
    float* zn = out + N_F;
    zn[(row << 2) + 0] = (float)i0;
    zn[(row << 2) + 1] = (float)i1;
    zn[(row << 2) + 2] = (float)i2;
    zn[(row << 2) + 3] = (float)i3;
}

extern "C" void kernel_launch(void* const* d_in, const int* in_sizes, int n_in,
                              void* d_out, int out_size, void* d_ws, size_t ws_size,
                              hipStream_t stream) {
    const float* x_f = (const float*)d_in[0];
    float* out = (float*)d_out;
    allen_cahn_knn_inv_kernel<<<GRID, BLOCK, 0, stream>>>(x_f, out);
}